// GAT_652835029727
// MI455X (gfx1250) — compile-verified
//
#include <hip/hip_runtime.h>
#include <hip/hip_bf16.h>

// Sizes from the reference
#define NN      50000
#define EE      800000
#define IN_DIM  128
#define HID     32
#define HEADS   4
#define H1DIM   128      // HID*HEADS
#define OUTD    64
#define BB      512
#define MLPH    1024

typedef float v2f __attribute__((ext_vector_type(2)));
typedef float v8f __attribute__((ext_vector_type(8)));

// ---------------------------------------------------------------------------
// fp32 WMMA: D(16x16) = A(16x4) x B(4x16) + C   (V_WMMA_F32_16X16X4_F32)
// Layouts (CDNA5 ISA 7.12.2):
//   A: lane L -> row (L&15), K = r + 2*(L>>4)   (contiguous float2)
//   B: lane L -> col (L&15), K = r + 2*(L>>4)   (contiguous float2 of W row)
//   C/D: lane L -> col (L&15), row = r + 8*(L>>4)
// ---------------------------------------------------------------------------
static __device__ inline v8f wmma_f32_16x16x4(v2f a, v2f b, v8f c) {
#if __has_builtin(__builtin_amdgcn_wmma_f32_16x16x4_f32)
    return __builtin_amdgcn_wmma_f32_16x16x4_f32(false, a, false, b, (short)0, c,
                                                 false, false);
#else
    // Should not trigger on gfx1250 toolchains (probe: builtin declared, 8 args).
    c[0] += a[0] * b[0];
    return c;
#endif
}

// out[M,NC] = A[M,K] * W[NC,K]^T  (+bias, +relu per FLAGS: 1=bias, 2=relu)
// One wave per 16x16 output tile; 4 waves (128 threads) per block.
template <int FLAGS>
__global__ __launch_bounds__(128) void wmma_gemm_k(const float* __restrict__ A,
                                                   const float* __restrict__ W,
                                                   const float* __restrict__ bias,
                                                   float* __restrict__ out,
                                                   int M, int K, int NC) {
    const int wave = threadIdx.x >> 5;
    const int lane = threadIdx.x & 31;
    const int mt   = blockIdx.x * 4 + wave;   // tile row
    if (mt * 16 >= M) return;                 // wave-uniform exit (EXEC stays all-1 for WMMA)
    const int nt = blockIdx.y;                // tile col
    const int li = lane & 15;
    const int hi = lane >> 4;

    const float* Ar = A + (size_t)(mt * 16 + li) * K + 2 * hi;
    const float* Wr = W + (size_t)(nt * 16 + li) * K + 2 * hi;

    v8f acc;
    float bv = 0.0f;
    if (FLAGS & 1) bv = bias[nt * 16 + li];   // per-column bias (all acc rows share col li)
#pragma unroll
    for (int r = 0; r < 8; ++r) acc[r] = bv;

#pragma unroll 4
    for (int k = 0; k < K; k += 4) {
        v2f a = *(const v2f*)(Ar + k);
        v2f b = *(const v2f*)(Wr + k);
        acc = wmma_f32_16x16x4(a, b, acc);
    }

    float* o = out + (size_t)(mt * 16 + 8 * hi) * NC + nt * 16 + li;
#pragma unroll
    for (int r = 0; r < 8; ++r) {
        float v = acc[r];
        if (FLAGS & 2) v = fmaxf(v, 0.0f);
        o[(size_t)r * NC] = v;
    }
}

// ---------------------------------------------------------------------------
// Edge scatter: out[row[e], :] += z[col[e], :]   (alpha==1 in the reference)
// F/4 float4 chunks per edge; consecutive lanes cover one edge row -> coalesced
// gathers. Prefetch a future edge's source row into L2 (global_prefetch_b8).
// ---------------------------------------------------------------------------
template <int F>
__global__ __launch_bounds__(256) void scatter_add_k(const float* __restrict__ z,
                                                     const int* __restrict__ ei,
                                                     float* __restrict__ out, int E) {
    const int CPE = F / 4;
    int t = blockIdx.x * 256 + threadIdx.x;
    if (t >= E * CPE) return;
    int e = t / CPE;
    int c = (t % CPE) * 4;
    int dst = ei[e];        // edge_index[0, e]
    int src = ei[E + e];    // edge_index[1, e]

    int e2 = e + 2048;      // hide random-gather latency
    if (e2 < E) {
        int s2 = ei[E + e2];
        __builtin_prefetch(z + (size_t)s2 * F + c, 0, 0);
    }

    const float4 v = *(const float4*)(z + (size_t)src * F + c);
    float* o = out + (size_t)dst * F + c;
    atomicAdd(o + 0, v.x);
    atomicAdd(o + 1, v.y);
    atomicAdd(o + 2, v.z);
    atomicAdd(o + 3, v.w);
}

__global__ __launch_bounds__(256) void elu_k(float* __restrict__ h, int n) {
    int t = blockIdx.x * 256 + threadIdx.x;
    if (t >= n) return;
    float v = h[t];
    h[t] = v > 0.0f ? v : (expf(v) - 1.0f);
}

// Global mean pool, phase 1: segment sums + counts
__global__ __launch_bounds__(256) void pool_sum_k(const float* __restrict__ h,
                                                  const int* __restrict__ batch,
                                                  float* __restrict__ g,
                                                  float* __restrict__ cnt) {
    int t = blockIdx.x * 256 + threadIdx.x;
    if (t >= NN * OUTD) return;
    int n = t >> 6;          // OUTD == 64
    int c = t & 63;
    int b = batch[n];
    atomicAdd(&g[b * OUTD + c], h[t]);
    if (c == 0) atomicAdd(&cnt[b], 1.0f);
}

__global__ __launch_bounds__(256) void pool_div_k(float* __restrict__ g,
                                                  const float* __restrict__ cnt) {
    int t = blockIdx.x * 256 + threadIdx.x;
    if (t >= BB * OUTD) return;
    g[t] /= fmaxf(cnt[t >> 6], 1.0f);
}

// Final layer [B,1024] -> [B,4] + softmax. One wave per graph row.
__global__ __launch_bounds__(32) void head_k(const float* __restrict__ a,
                                             const float* __restrict__ w,
                                             const float* __restrict__ bias,
                                             float* __restrict__ out) {
    int b = blockIdx.x;
    int lane = threadIdx.x;
    const float* ar = a + (size_t)b * MLPH;
    float s0 = 0.f, s1 = 0.f, s2 = 0.f, s3 = 0.f;
    for (int k = lane; k < MLPH; k += 32) {
        float av = ar[k];
        s0 = fmaf(av, w[k], s0);
        s1 = fmaf(av, w[MLPH + k], s1);
        s2 = fmaf(av, w[2 * MLPH + k], s2);
        s3 = fmaf(av, w[3 * MLPH + k], s3);
    }
#pragma unroll
    for (int off = 16; off > 0; off >>= 1) {
        s0 += __shfl_down(s0, off);
        s1 += __shfl_down(s1, off);
        s2 += __shfl_down(s2, off);
        s3 += __shfl_down(s3, off);
    }
    if (lane == 0) {
        float l0 = s0 + bias[0], l1 = s1 + bias[1];
        float l2 = s2 + bias[2], l3 = s3 + bias[3];
        float m = fmaxf(fmaxf(l0, l1), fmaxf(l2, l3));
        float e0 = expf(l0 - m), e1 = expf(l1 - m);
        float e2 = expf(l2 - m), e3 = expf(l3 - m);
        float inv = 1.0f / (e0 + e1 + e2 + e3);
        out[b * 4 + 0] = e0 * inv;
        out[b * 4 + 1] = e1 * inv;
        out[b * 4 + 2] = e2 * inv;
        out[b * 4 + 3] = e3 * inv;
    }
}

// ---------------------------------------------------------------------------
extern "C" void kernel_launch(void* const* d_in, const int* in_sizes, int n_in,
                              void* d_out, int out_size, void* d_ws, size_t ws_size,
                              hipStream_t stream) {
    const float* x     = (const float*)d_in[0];
    const int*   ei    = (const int*)d_in[1];   // [2, E]
    const int*   batch = (const int*)d_in[2];
    const float* fc1   = (const float*)d_in[3]; // [H*HID, IN_DIM] flattened
    const float* fc2   = (const float*)d_in[5]; // [OUTD, H1DIM]
    const float* w1 = (const float*)d_in[7],  *b1 = (const float*)d_in[8];
    const float* w2 = (const float*)d_in[9],  *b2 = (const float*)d_in[10];
    const float* w3 = (const float*)d_in[11], *b3 = (const float*)d_in[12];
    const float* w4 = (const float*)d_in[13], *b4 = (const float*)d_in[14];
    const float* w5 = (const float*)d_in[15], *b5 = (const float*)d_in[16];
    // d_in[4], d_in[6] (attn weights) are dead: softmax over a size-1 axis == 1.

    float* ws = (float*)d_ws;
    size_t o = 0;
    float* z1  = ws + o; o += (size_t)NN * H1DIM;   // 6.4M
    float* z2  = ws + o; o += (size_t)NN * OUTD;    // 3.2M
    float* h1  = ws + o; o += (size_t)NN * H1DIM;   // 6.4M  (accum, zeroed)
    float* h2  = ws + o; o += (size_t)NN * OUTD;    // 3.2M  (accum, zeroed)
    float* g   = ws + o; o += (size_t)BB * OUTD;    //       (accum, zeroed)
    float* cnt = ws + o; o += (size_t)BB;           //       (accum, zeroed)
    float* a1  = ws + o; o += (size_t)BB * MLPH;
    float* a2  = ws + o; o += (size_t)BB * MLPH;

    // Zero the atomic accumulators (h1,h2,g,cnt are contiguous).
    size_t zero_bytes = ((size_t)NN * H1DIM + (size_t)NN * OUTD +
                         (size_t)BB * OUTD + (size_t)BB) * sizeof(float);
    hipMemsetAsync(h1, 0, zero_bytes, stream);

    const int MT_N = (NN / 16 + 3) / 4;   // 3125 row tiles / 4 waves per block
    // Layer 1 GEMM: z1 = x @ fc1^T  [N,128]x[128,128]
    wmma_gemm_k<0><<<dim3(MT_N, H1DIM / 16), 128, 0, stream>>>(x, fc1, nullptr, z1,
                                                               NN, IN_DIM, H1DIM);
    // Edge aggregation 1
    scatter_add_k<H1DIM><<<(EE * (H1DIM / 4) + 255) / 256, 256, 0, stream>>>(z1, ei, h1, EE);
    // ELU
    elu_k<<<((NN * H1DIM) + 255) / 256, 256, 0, stream>>>(h1, NN * H1DIM);
    // Layer 2 GEMM: z2 = h1 @ fc2^T  [N,128]x[128,64]
    wmma_gemm_k<0><<<dim3(MT_N, OUTD / 16), 128, 0, stream>>>(h1, fc2, nullptr, z2,
                                                              NN, H1DIM, OUTD);
    // Edge aggregation 2
    scatter_add_k<OUTD><<<(EE * (OUTD / 4) + 255) / 256, 256, 0, stream>>>(z2, ei, h2, EE);
    // Global mean pool
    pool_sum_k<<<(NN * OUTD + 255) / 256, 256, 0, stream>>>(h2, batch, g, cnt);
    pool_div_k<<<(BB * OUTD + 255) / 256, 256, 0, stream>>>(g, cnt);

    // MLP head (bias+relu fused into WMMA epilogue)
    const int MT_B = (BB / 16 + 3) / 4;   // 8 blocks of 4 waves = 32 row tiles
    wmma_gemm_k<3><<<dim3(MT_B, MLPH / 16), 128, 0, stream>>>(g, w1, b1, a1, BB, OUTD, MLPH);
    wmma_gemm_k<3><<<dim3(MT_B, MLPH / 16), 128, 0, stream>>>(a1, w2, b2, a2, BB, MLPH, MLPH);
    wmma_gemm_k<3><<<dim3(MT_B, MLPH / 16), 128, 0, stream>>>(a2, w3, b3, a1, BB, MLPH, MLPH);
    wmma_gemm_k<3><<<dim3(MT_B, MLPH / 16), 128, 0, stream>>>(a1, w4, b4, a2, BB, MLPH, MLPH);
    // Final 1024 -> 4 + softmax
    head_k<<<BB, 32, 0, stream>>>(a2, w5, b5, (float*)d_out);
    (void)in_sizes; (void)n_in; (void)out_size; (void)ws_size;
}